// sfr_64484638982577
// MI455X (gfx1250) — compile-verified
//
#include <hip/hip_runtime.h>

// ---------------------------------------------------------------------------
// Batched RK4 integration of r' = R e3, R' = R*u_hat (u_hat constant per elem).
// Bandwidth-bound: 161.5 MB of output -> ~7us floor at 23.3 TB/s.
// CDNA5 path: LDS-staged snapshots + GLOBAL_STORE_ASYNC_FROM_LDS_B64
// (ASYNCcnt-tracked, double-buffered) for fully coalesced streaming stores.
// ---------------------------------------------------------------------------

#if defined(__has_builtin)
#  if __has_builtin(__builtin_amdgcn_global_store_async_from_lds_b64) && \
      __has_builtin(__builtin_amdgcn_s_wait_asynccnt)
#    define SFR_USE_ASYNC 1
#  endif
#endif
#ifndef SFR_USE_ASYNC
#  define SFR_USE_ASYNC 0
#endif

#if SFR_USE_ASYNC
// Builtin signature (from clang diagnostic): param0 is a pointer to a 2-int
// vector in addrspace(1) (printed as "__device__"); param1 is the LDS side.
typedef __attribute__((ext_vector_type(2))) int v2i;
typedef __attribute__((address_space(1))) v2i g_v2i;  // global b64 chunk
typedef __attribute__((address_space(3))) v2i l_v2i;  // LDS b64 chunk
#endif

namespace {
constexpr int TPB   = 256;  // 8 wave32 per block
constexpr int NST   = 10;   // RK4 steps
constexpr int STATE = 14;   // floats per element
}

// dR = R(3x3) * u_hat, with u_hat = [[0,0,uy],[0,0,-ux],[-uy,ux,0]]
__device__ __forceinline__ void ode_dR(const float R[9], float ux, float uy,
                                       float dR[9]) {
#pragma unroll
  for (int i = 0; i < 3; ++i) {
    const float r0 = R[3 * i + 0], r1 = R[3 * i + 1], r2 = R[3 * i + 2];
    dR[3 * i + 0] = r2 * (-uy);
    dR[3 * i + 1] = r2 * ux;
    dR[3 * i + 2] = r0 * uy - r1 * ux;
  }
}

__global__ __launch_bounds__(TPB) void sfr_rk4_kernel(
    const float* __restrict__ actions, float* __restrict__ out, int B) {
  __shared__ float stage[2][TPB * STATE];  // double-buffered snapshot staging
  const int tid = threadIdx.x;
  const int b   = blockIdx.x * TPB + tid;

  const float lD = (float)(0.05 * 0.0075);  // l_top * D = 3.75e-4
  const float a1 = actions[3 * b + 1];
  const float a2 = actions[3 * b + 2];
  const float ux = a2 / (-lD);
  const float uy = a1 / lD;

  const float dt  = 0.005f;
  const float hdt = 0.5f * dt;
  const float dt6 = dt / 6.0f;

  float r[3] = {0.f, 0.f, 0.f};
  float R[9] = {1.f, 0.f, 0.f, 0.f, 1.f, 0.f, 0.f, 0.f, 1.f};

  auto emit = [&](int t) {
    const int buf = t & 1;
#if SFR_USE_ASYNC
    // Buffer (t&1) was last used by snapshot t-2; its 7 per-wave async stores
    // must have retired (stores complete in order -> waiting to <=7 leaves
    // only snapshot t-1 in flight).
    if (t >= 2) __builtin_amdgcn_s_wait_asynccnt(7);
    __syncthreads();
    float* row = &stage[buf][tid * STATE];
    row[0] = r[0]; row[1] = r[1]; row[2] = r[2];
#pragma unroll
    for (int k = 0; k < 9; ++k) row[3 + k] = R[k];
    row[12] = ux; row[13] = uy;
    __syncthreads();
    // Block's snapshot is 256*14 floats, contiguous in global [t][b][14].
    char* gbase = (char*)out +
        (((size_t)t * (size_t)B + (size_t)blockIdx.x * TPB) * STATE) * sizeof(float);
    char* lbase = (char*)&stage[buf][0];
#pragma unroll
    for (int c = 0; c < 7; ++c) {
      const int off = (tid + c * TPB) * 8;  // lane-contiguous 8B chunks
      __builtin_amdgcn_global_store_async_from_lds_b64(
          (g_v2i*)(gbase + off), (l_v2i*)(lbase + off), 0, 0);
    }
#else
    (void)buf;
    float y[STATE] = {r[0], r[1], r[2], R[0], R[1], R[2], R[3], R[4],
                      R[5], R[6], R[7], R[8], ux,  uy};
    float2* dst = (float2*)(out + ((size_t)t * (size_t)B + (size_t)b) * STATE);
#pragma unroll
    for (int k = 0; k < 7; ++k) dst[k] = make_float2(y[2 * k], y[2 * k + 1]);
#endif
  };

  emit(0);

#pragma unroll 1
  for (int s = 0; s < NST; ++s) {
    float k1[9], k2[9], k3[9], k4[9], T[9];

    ode_dR(R, ux, uy, k1);
    const float k1r0 = R[2], k1r1 = R[5], k1r2 = R[8];
#pragma unroll
    for (int i = 0; i < 9; ++i) T[i] = R[i] + hdt * k1[i];

    ode_dR(T, ux, uy, k2);
    const float k2r0 = T[2], k2r1 = T[5], k2r2 = T[8];
#pragma unroll
    for (int i = 0; i < 9; ++i) T[i] = R[i] + hdt * k2[i];

    ode_dR(T, ux, uy, k3);
    const float k3r0 = T[2], k3r1 = T[5], k3r2 = T[8];
#pragma unroll
    for (int i = 0; i < 9; ++i) T[i] = R[i] + dt * k3[i];

    ode_dR(T, ux, uy, k4);
    const float k4r0 = T[2], k4r1 = T[5], k4r2 = T[8];

#pragma unroll
    for (int i = 0; i < 9; ++i)
      R[i] = R[i] + dt6 * (((k1[i] + 2.0f * k2[i]) + 2.0f * k3[i]) + k4[i]);
    r[0] = r[0] + dt6 * (((k1r0 + 2.0f * k2r0) + 2.0f * k3r0) + k4r0);
    r[1] = r[1] + dt6 * (((k1r1 + 2.0f * k2r1) + 2.0f * k3r1) + k4r1);
    r[2] = r[2] + dt6 * (((k1r2 + 2.0f * k2r2) + 2.0f * k3r2) + k4r2);

    emit(s + 1);
  }
}

extern "C" void kernel_launch(void* const* d_in, const int* in_sizes, int n_in,
                              void* d_out, int out_size, void* d_ws,
                              size_t ws_size, hipStream_t stream) {
  const float* actions = (const float*)d_in[0];
  float* out = (float*)d_out;
  const int B = in_sizes[0] / 3;  // BATCH = 262144, a multiple of 256
  const int grid = B / TPB;
  sfr_rk4_kernel<<<grid, TPB, 0, stream>>>(actions, out, B);
  (void)n_in; (void)out_size; (void)d_ws; (void)ws_size;
}